// GCN_36636071034924
// MI455X (gfx1250) — compile-verified
//
#include <hip/hip_runtime.h>
#include <math.h>

typedef __attribute__((ext_vector_type(2))) float v2f;
typedef __attribute__((ext_vector_type(8))) float v8f;

// ---------------------------------------------------------------------------
// Degree / normalization kernels
// ---------------------------------------------------------------------------
__global__ void deg_init_kernel(float* __restrict__ deg, int n) {
    int i = blockIdx.x * blockDim.x + threadIdx.x;
    if (i < n) deg[i] = 1.0f;   // self-loop contributes 1 to every node
}

__global__ void deg_count_kernel(const long long* __restrict__ dst,
                                 float* __restrict__ deg, int E) {
    int e = blockIdx.x * blockDim.x + threadIdx.x;
    if (e < E) atomicAdd(&deg[(int)dst[e]], 1.0f);
}

__global__ void dinv_kernel(const float* __restrict__ deg,
                            float* __restrict__ dinv, int n) {
    int i = blockIdx.x * blockDim.x + threadIdx.x;
    if (i < n) {
        float d = deg[i];
        dinv[i] = (d > 0.0f) ? rsqrtf(d) : 0.0f;
    }
}

// ---------------------------------------------------------------------------
// WMMA f32 GEMM:  C[16 x 32 tile] = A[16 x K] * W[K x 32]  per block (2 waves)
//   A row-major [N x K], W row-major [K x ldw], C row-major [rows x ldc].
//   Store guarded to col < ncols; optional fused bias (classifier).
// ---------------------------------------------------------------------------
template <int K>
__global__ __launch_bounds__(64)
void wmma_gemm_kernel(const float* __restrict__ A, const float* __restrict__ W,
                      float* __restrict__ C, const float* __restrict__ bias,
                      int ldw, int ldc, int ncols, int addBias)
{
    __shared__ __align__(16) float sA[16 * K];   // 16 rows x K
    __shared__ __align__(16) float sW[K * 32];   // K rows x 32 cols

    const int rb  = blockIdx.x * 16;
    const int cb  = blockIdx.y * 32;
    const int tid = threadIdx.x;

    // Stage A tile (float4, fully coalesced; K is a multiple of 4)
    for (int idx = tid; idx < 16 * (K / 4); idx += 64) {
        int r = idx / (K / 4);
        int q = idx % (K / 4);
        float4 v = *(const float4*)(A + (size_t)(rb + r) * K + q * 4);
        *(float4*)(sA + r * K + q * 4) = v;
    }
    // Stage W tile (K x 32)
    for (int idx = tid; idx < K * 8; idx += 64) {
        int r = idx / 8;
        int q = idx % 8;
        float4 v = *(const float4*)(W + (size_t)r * ldw + cb + q * 4);
        *(float4*)(sW + r * 32 + q * 4) = v;
    }
    __syncthreads();

    const int lane = tid & 31;
    const int wave = tid >> 5;          // which 16-col sub-tile
    const int half = lane >> 4;         // K-pair selector per ISA A/B layout
    const int m    = lane & 15;
    const int cloc = wave * 16 + m;     // local column 0..31

    v8f acc = {};
    #pragma unroll 4
    for (int kk = 0; kk < K; kk += 4) {
        const int k0 = kk + 2 * half;
        v2f a, b;
        a.x = sA[m * K + k0];
        a.y = sA[m * K + k0 + 1];
        b.x = sW[k0 * 32 + cloc];
        b.y = sW[(k0 + 1) * 32 + cloc];
        acc = __builtin_amdgcn_wmma_f32_16x16x4_f32(
                  false, a, false, b, (short)0, acc, false, false);
    }

    const int col = cb + cloc;
    if (col < ncols) {
        const float bv = addBias ? bias[col] : 0.0f;
        #pragma unroll
        for (int r = 0; r < 8; ++r) {
            // D layout: VGPR r holds rows (r) for lanes 0-15, (r+8) for 16-31
            const int row = rb + r + 8 * half;
            C[(size_t)row * ldc + col] = acc[r] + bv;
        }
    }
}

// ---------------------------------------------------------------------------
// Edge scatter: agg[dst] += t[src] * dinv[src]*dinv[dst]
// one thread per (edge, float4 chunk) -> coalesced gathers, f32 atomics
// ---------------------------------------------------------------------------
__global__ void scatter_kernel(const float* __restrict__ t,
                               const long long* __restrict__ src,
                               const long long* __restrict__ dst,
                               const float* __restrict__ dinv,
                               float* __restrict__ agg, int E, int F)
{
    const long long idx = (long long)blockIdx.x * blockDim.x + threadIdx.x;
    const int chunks = F >> 2;
    const long long e = idx / chunks;
    const int c = (int)(idx % chunks);
    if (e >= E) return;
    const int s = (int)src[e];
    const int d = (int)dst[e];
    const float nrm = dinv[s] * dinv[d];
    float4 v = *(const float4*)(t + (size_t)s * F + c * 4);
    float* p = agg + (size_t)d * F + c * 4;
    atomicAdd(p + 0, v.x * nrm);
    atomicAdd(p + 1, v.y * nrm);
    atomicAdd(p + 2, v.z * nrm);
    atomicAdd(p + 3, v.w * nrm);
}

// ---------------------------------------------------------------------------
// h = relu(agg + t * dinv^2 + bias)   (self-loop folded in analytically)
// ---------------------------------------------------------------------------
__global__ void self_bias_relu_kernel(float* __restrict__ h,
                                      const float* __restrict__ t,
                                      const float* __restrict__ dinv,
                                      const float* __restrict__ bias,
                                      int N, int F)
{
    const long long idx = (long long)blockIdx.x * blockDim.x + threadIdx.x;
    const int chunks = F >> 2;
    if (idx >= (long long)N * chunks) return;
    const int i = (int)(idx / chunks);
    const int c = (int)(idx % chunks);
    const float di = dinv[i];
    const float sn = di * di;
    float4 a  = *(float4*)(h + (size_t)i * F + c * 4);
    float4 tv = *(const float4*)(t + (size_t)i * F + c * 4);
    float4 b  = *(const float4*)(bias + c * 4);
    a.x = fmaxf(a.x + tv.x * sn + b.x, 0.0f);
    a.y = fmaxf(a.y + tv.y * sn + b.y, 0.0f);
    a.z = fmaxf(a.z + tv.z * sn + b.z, 0.0f);
    a.w = fmaxf(a.w + tv.w * sn + b.w, 0.0f);
    *(float4*)(h + (size_t)i * F + c * 4) = a;
}

// ---------------------------------------------------------------------------
// Zero-pad classifier weight [K x ncls] -> [K x ldp]
// ---------------------------------------------------------------------------
__global__ void pad_w_kernel(const float* __restrict__ Wsrc, float* __restrict__ Wpad,
                             int K, int ncls, int ldp)
{
    int idx = blockIdx.x * blockDim.x + threadIdx.x;
    if (idx >= K * ldp) return;
    int r = idx / ldp;
    int c = idx % ldp;
    Wpad[idx] = (c < ncls) ? Wsrc[r * ncls + c] : 0.0f;
}

// ---------------------------------------------------------------------------
extern "C" void kernel_launch(void* const* d_in, const int* in_sizes, int n_in,
                              void* d_out, int out_size, void* d_ws, size_t ws_size,
                              hipStream_t stream)
{
    const float*     X   = (const float*)d_in[0];
    const long long* ei  = (const long long*)d_in[1];
    const float*     W1  = (const float*)d_in[2];
    const float*     b1  = (const float*)d_in[3];
    const float*     W2  = (const float*)d_in[4];
    const float*     b2  = (const float*)d_in[5];
    const float*     Wc  = (const float*)d_in[6];
    const float*     bc  = (const float*)d_in[7];
    float*           out = (float*)d_out;

    const int h1     = in_sizes[3];            // 256
    const int in_dim = in_sizes[2] / h1;       // 128
    const int h2     = in_sizes[5];            // 128
    const int ncls   = in_sizes[7];            // 40
    const int N      = in_sizes[0] / in_dim;   // 50000
    const int E      = in_sizes[1] / 2;        // 640000
    const int NCLS_PAD = 64;

    const long long* src = ei;
    const long long* dst = ei + E;

    // Workspace layout
    auto al = [](size_t x) { return (x + 255) & ~(size_t)255; };
    char* ws = (char*)d_ws;
    size_t szT   = al((size_t)N * h1 * sizeof(float));   // t buffer (max width h1)
    size_t szH   = al((size_t)N * h1 * sizeof(float));   // agg / h buffer
    size_t szDeg = al((size_t)N * sizeof(float));
    float* tbuf = (float*)(ws);
    float* hbuf = (float*)(ws + szT);
    float* deg  = (float*)(ws + szT + szH);
    float* dinv = (float*)(ws + szT + szH + szDeg);
    float* Wcp  = (float*)(ws + szT + szH + 2 * szDeg);

    // 1) degrees + normalization
    deg_init_kernel<<<(N + 255) / 256, 256, 0, stream>>>(deg, N);
    deg_count_kernel<<<(E + 255) / 256, 256, 0, stream>>>(dst, deg, E);
    dinv_kernel<<<(N + 255) / 256, 256, 0, stream>>>(deg, dinv, N);

    // 2) t1 = X @ W1   (K=128, out 256)
    wmma_gemm_kernel<128><<<dim3(N / 16, h1 / 32), 64, 0, stream>>>(
        X, W1, tbuf, nullptr, h1, h1, h1, 0);

    // 3) agg1 = scatter(t1)
    hipMemsetAsync(hbuf, 0, (size_t)N * h1 * sizeof(float), stream);
    {
        long long total = (long long)E * (h1 / 4);
        scatter_kernel<<<(unsigned)((total + 255) / 256), 256, 0, stream>>>(
            tbuf, src, dst, dinv, hbuf, E, h1);
    }
    // 4) h1 = relu(agg1 + t1*dinv^2 + b1)
    {
        long long total = (long long)N * (h1 / 4);
        self_bias_relu_kernel<<<(unsigned)((total + 255) / 256), 256, 0, stream>>>(
            hbuf, tbuf, dinv, b1, N, h1);
    }

    // 5) t2 = h1 @ W2   (K=256, out 128) -> reuse tbuf
    wmma_gemm_kernel<256><<<dim3(N / 16, h2 / 32), 64, 0, stream>>>(
        hbuf, W2, tbuf, nullptr, h2, h2, h2, 0);

    // 6) agg2 = scatter(t2) -> reuse hbuf (stream-ordered after GEMM2)
    hipMemsetAsync(hbuf, 0, (size_t)N * h2 * sizeof(float), stream);
    {
        long long total = (long long)E * (h2 / 4);
        scatter_kernel<<<(unsigned)((total + 255) / 256), 256, 0, stream>>>(
            tbuf, src, dst, dinv, hbuf, E, h2);
    }
    // 7) h2 = relu(agg2 + t2*dinv^2 + b2)
    {
        long long total = (long long)N * (h2 / 4);
        self_bias_relu_kernel<<<(unsigned)((total + 255) / 256), 256, 0, stream>>>(
            hbuf, tbuf, dinv, b2, N, h2);
    }

    // 8) logits = h2 @ Wc + bc   (pad Wc to 64 cols, guard stores to 40)
    pad_w_kernel<<<(h2 * NCLS_PAD + 255) / 256, 256, 0, stream>>>(
        Wc, Wcp, h2, ncls, NCLS_PAD);
    wmma_gemm_kernel<128><<<dim3(N / 16, NCLS_PAD / 32), 64, 0, stream>>>(
        hbuf, Wcp, out, bc, NCLS_PAD, ncls, ncls, 1);
}